// Loss_func_14044543058257
// MI455X (gfx1250) — compile-verified
//
#include <hip/hip_runtime.h>
#include <math.h>

// Shapes: B=256, L=4, K=16, Ml=64, N=4, M=4
//   p:     [256, 64, 64]           f32
//   Theta: [256, 4, 64, 2]         f32
//   Tx:    [256, 4, 16, 64, 4, 2]  f32
//   Rx:    [256, 4, 16, 4, 64, 2]  f32
//   sigma2:[1]                     f32
// out:     scalar f32 (mean rate)

typedef float v2f __attribute__((ext_vector_type(2)));
typedef float v8f __attribute__((ext_vector_type(8)));

#define LDA 33   // padded stride for 64x32 A chunks (RT)
#define LDB 65   // padded stride for 32x64 B chunks (Tx)

__device__ __forceinline__ v8f wmma_f32(v2f a, v2f b, v8f c) {
  // D = A(16x4,f32) * B(4x16,f32) + C(16x16,f32)
  return __builtin_amdgcn_wmma_f32_16x16x4_f32(
      /*neg_a=*/false, a, /*neg_b=*/false, b,
      /*c_mod=*/(short)0, c, /*reuse_a=*/false, /*reuse_b=*/false);
}

__global__ __launch_bounds__(256)
void batch_rate_kernel(const float* __restrict__ p,
                       const float* __restrict__ Theta,
                       const float* __restrict__ Tx,
                       const float* __restrict__ Rx,
                       const float* __restrict__ sigma2,
                       float* __restrict__ rates) {
  __shared__ float ldsAr[64 * LDA];
  __shared__ float ldsAi[64 * LDA];
  __shared__ float ldsBr[32 * LDB];
  __shared__ float ldsBi[32 * LDB];
  __shared__ float rowsum[64];
  __shared__ float diagv[64];
  __shared__ float terms[64];

  const int b    = blockIdx.x;       // batch
  const int tid  = threadIdx.x;      // 0..255 (8 wave32s)
  const int lane = tid & 31;
  const int wave = tid >> 5;
  const int lhalf = lane >> 4;       // 0: lanes 0-15, 1: lanes 16-31
  const int lr    = lane & 15;

  // Each wave owns tiles t0=2*wave and t0+1 of the 4x4 tile grid of h[64x64].
  // Both tiles share the same row block ti.
  const int t0  = wave * 2;
  const int ti  = t0 >> 2;
  const int tj0 = t0 & 3;
  const int tj1 = (t0 + 1) & 3;

  v8f accr[2], acci[2];
#pragma unroll
  for (int t = 0; t < 2; ++t) {
    accr[t] = (v8f){0.f, 0.f, 0.f, 0.f, 0.f, 0.f, 0.f, 0.f};
    acci[t] = (v8f){0.f, 0.f, 0.f, 0.f, 0.f, 0.f, 0.f, 0.f};
  }

  const float2* Tx2 = reinterpret_cast<const float2*>(Tx);
  const float2* Rx2 = reinterpret_cast<const float2*>(Rx);
  const float2* Th2 = reinterpret_cast<const float2*>(Theta);

  // K = 256 reduction, chunks of 32 staged through LDS.
  for (int kc = 0; kc < 256; kc += 32) {
    // Stage B chunk: ldsB[kk][j] = Tx_m[b, kc+kk, j]  (complex)
    for (int e = tid; e < 32 * 64; e += 256) {
      const int kk = e >> 6, j = e & 63;
      const int r  = kc + kk;
      const int l  = r >> 6, ml = r & 63;
      const int k  = j >> 2, n  = j & 3;
      const float2 v = Tx2[(((b * 4 + l) * 16 + k) * 64 + ml) * 4 + n];
      ldsBr[kk * LDB + j] = v.x;
      ldsBi[kk * LDB + j] = v.y;
    }
    // Stage A chunk: ldsA[i][kk] = (Rx_m * Theta)[b, i, kc+kk]  (complex scale fused)
    for (int e = tid; e < 64 * 32; e += 256) {
      const int i  = e >> 5, kk = e & 31;
      const int r  = kc + kk;
      const int l  = r >> 6, ml = r & 63;
      const int k  = i >> 2, m  = i & 3;
      const float2 rx = Rx2[(((b * 4 + l) * 16 + k) * 4 + m) * 64 + ml];
      const float2 th = Th2[(b * 4 + l) * 64 + ml];
      ldsAr[i * LDA + kk] = rx.x * th.x - rx.y * th.y;
      ldsAi[i * LDA + kk] = rx.x * th.y + rx.y * th.x;
    }
    __syncthreads();

    const int arow  = ti * 16 + lr;
    const int bcol0 = tj0 * 16 + lr;
    const int bcol1 = tj1 * 16 + lr;

#pragma unroll
    for (int ks = 0; ks < 32; ks += 4) {
      const int ka = ks + 2 * lhalf;  // lanes 0-15: K=ks..ks+1; 16-31: K=ks+2..ks+3
      v2f ar, ai, ain;
      ar.x = ldsAr[arow * LDA + ka];
      ar.y = ldsAr[arow * LDA + ka + 1];
      ai.x = ldsAi[arow * LDA + ka];
      ai.y = ldsAi[arow * LDA + ka + 1];
      ain.x = -ai.x;
      ain.y = -ai.y;

      v2f br0, bi0, br1, bi1;
      br0.x = ldsBr[ka * LDB + bcol0]; br0.y = ldsBr[(ka + 1) * LDB + bcol0];
      bi0.x = ldsBi[ka * LDB + bcol0]; bi0.y = ldsBi[(ka + 1) * LDB + bcol0];
      br1.x = ldsBr[ka * LDB + bcol1]; br1.y = ldsBr[(ka + 1) * LDB + bcol1];
      bi1.x = ldsBi[ka * LDB + bcol1]; bi1.y = ldsBi[(ka + 1) * LDB + bcol1];

      // h_r += Ar*Br - Ai*Bi ; h_i += Ar*Bi + Ai*Br   (4 WMMAs per tile)
      accr[0] = wmma_f32(ar,  br0, accr[0]);
      accr[0] = wmma_f32(ain, bi0, accr[0]);
      acci[0] = wmma_f32(ar,  bi0, acci[0]);
      acci[0] = wmma_f32(ai,  br0, acci[0]);

      accr[1] = wmma_f32(ar,  br1, accr[1]);
      accr[1] = wmma_f32(ain, bi1, accr[1]);
      acci[1] = wmma_f32(ar,  bi1, acci[1]);
      acci[1] = wmma_f32(ai,  br1, acci[1]);
    }
    __syncthreads();
  }

  // Epilogue: h_gain = p * |h|^2; row sums + diagonal.
  for (int i = tid; i < 64; i += 256) {
    rowsum[i] = 0.f;
    diagv[i]  = 0.f;
  }
  __syncthreads();

  const float* pb = p + b * 64 * 64;
#pragma unroll
  for (int t = 0; t < 2; ++t) {
    const int tj = (t == 0) ? tj0 : tj1;
#pragma unroll
    for (int v = 0; v < 8; ++v) {
      // C/D layout: VGPR v, lanes 0-15 -> row v, lanes 16-31 -> row v+8; col = lane%16
      const int row = ti * 16 + v + 8 * lhalf;
      const int col = tj * 16 + lr;
      const float hr = accr[t][v];
      const float hq = acci[t][v];
      const float g  = pb[row * 64 + col] * (hr * hr + hq * hq);
      atomicAdd(&rowsum[row], g);                    // ds_add_f32
      if (ti == tj && lr == v + 8 * lhalf) diagv[row] = g;  // exactly one writer
    }
  }
  __syncthreads();

  if (tid < 64) {
    const float s2  = sigma2[0];
    const float num = diagv[tid];
    const float den = rowsum[tid] - num + s2;
    terms[tid] = log1pf(num / den);
  }
  __syncthreads();

  if (tid == 0) {
    float s = 0.f;
    for (int i = 0; i < 64; ++i) s += terms[i];
    rates[b] = s * 1.44269504088896340736f;  // / ln(2)
  }
}

__global__ __launch_bounds__(256)
void reduce_mean_kernel(const float* __restrict__ rates, float* __restrict__ out) {
  __shared__ float sm[256];
  sm[threadIdx.x] = rates[threadIdx.x];
  __syncthreads();
  for (int s = 128; s > 0; s >>= 1) {
    if (threadIdx.x < s) sm[threadIdx.x] += sm[threadIdx.x + s];
    __syncthreads();
  }
  if (threadIdx.x == 0) out[0] = sm[0] * (1.0f / 256.0f);
}

extern "C" void kernel_launch(void* const* d_in, const int* in_sizes, int n_in,
                              void* d_out, int out_size, void* d_ws, size_t ws_size,
                              hipStream_t stream) {
  const float* p      = (const float*)d_in[0];
  const float* Theta  = (const float*)d_in[1];
  const float* Tx     = (const float*)d_in[2];
  const float* Rx     = (const float*)d_in[3];
  const float* sigma2 = (const float*)d_in[4];
  float* rates = (float*)d_ws;   // 256 floats of scratch
  float* out   = (float*)d_out;  // 1 float

  batch_rate_kernel<<<dim3(256), dim3(256), 0, stream>>>(p, Theta, Tx, Rx, sigma2, rates);
  reduce_mean_kernel<<<dim3(1), dim3(256), 0, stream>>>(rates, out);
}